// NeighConsensus_15719580303921
// MI455X (gfx1250) — compile-verified
//
#include <hip/hip_runtime.h>

typedef float v2f __attribute__((ext_vector_type(2)));
typedef float v8f __attribute__((ext_vector_type(8)));

#define SDIM 32
#define COUT 10

// ---------------------------------------------------------------------------
// Pack conv4d weights (cout,cin,3,3,3,3) into B-fragment order for
// V_WMMA_F32_16X16X4_F32:  dst[row][j2][n][p]  (row = ci*27 + dh1*9 + dw1*3 + dh2,
// K-slot within row = dw2 = 2*j2 + p, padded: dw2==3 -> 0, n>=COUT -> 0).
// permuted=1 builds W'(a,b,c,d) = W(c,d,a,b)  (the symmetric-pass kernel).
// ---------------------------------------------------------------------------
__global__ __launch_bounds__(256)
void pack_weights_kernel(const float* __restrict__ W, float* __restrict__ dst,
                         int cin, int permuted)
{
    int idx = blockIdx.x * 256 + threadIdx.x;
    int total = cin * 27 * 64;
    if (idx >= total) return;
    int p  = idx & 1;
    int n  = (idx >> 1) & 15;
    int j2 = (idx >> 5) & 1;
    int r  = idx >> 6;
    int ci = r / 27;
    int q  = r % 27;
    int a = q / 9, b = (q / 3) % 3, c = q % 3, d = 2 * j2 + p;
    float v = 0.0f;
    if (n < COUT && d < 3) {
        int t = permuted ? (c * 27 + d * 9 + a * 3 + b)
                         : (a * 27 + b * 9 + c * 3 + d);
        v = W[(n * cin + ci) * 81 + t];
    }
    dst[idx] = v;
}

// ---------------------------------------------------------------------------
// conv4d (SAME, 3^4 taps) + bias + ReLU via V_WMMA_F32_16X16X4_F32.
// Block = 128 threads (4 wave32). Block owns one (n,h1,w1) and 2 h2 lines.
// Wave tile: M = 16 consecutive w2, N = 16 couts (10 live), K = CIN*27*4.
// ---------------------------------------------------------------------------
template<int CIN>
__global__ __launch_bounds__(128)
void conv4d_wmma_kernel(const float* __restrict__ in,    // [4][CIN][32][32][32][32]
                        const float* __restrict__ bpk,   // [CIN*27][2][16][2]
                        const float* __restrict__ bias,  // [COUT]
                        float* __restrict__ out)         // [4][COUT][32][32][32][32]
{
    constexpr int H2E  = 4;    // h2 tile (2) + halo (2)
    constexpr int WROW = 36;   // w2 (32) + halo (2), padded
    __shared__ float raw[CIN * 9 * H2E * WROW];

    const int tid = threadIdx.x;
    const unsigned bb = blockIdx.x;
    const int n   = bb >> 14;
    const int h1  = (bb >> 9) & 31;
    const int w1  = (bb >> 4) & 31;
    const int h2b = (bb & 15) * 2;

    // ---- Phase 1: cooperative im2col halo tile -> LDS (zero-padded) ----
    constexpr int E = CIN * 9 * H2E * WROW;
    for (int idx = tid; idx < E; idx += 128) {
        int i   = idx % WROW;
        int t   = idx / WROW;
        int h2l = t & 3;  t >>= 2;
        int dw1 = t % 3;  t /= 3;
        int dh1 = t % 3;
        int ci  = t / 3;
        int h1g = h1 + dh1 - 1;
        int w1g = w1 + dw1 - 1;
        int h2g = h2b + h2l - 1;
        int w2g = i - 1;
        float v = 0.0f;
        if ((unsigned)h1g < 32u && (unsigned)w1g < 32u &&
            (unsigned)h2g < 32u && (unsigned)w2g < 32u) {
            v = in[(((((size_t)n * CIN + ci) * SDIM + h1g) * SDIM + w1g) * SDIM + h2g)
                   * SDIM + w2g];
        }
        raw[idx] = v;
    }
    __syncthreads();

    // ---- Phase 2: WMMA K-loop (EXEC all-ones, no divergence) ----
    const int lane = tid & 31;
    const int wave = tid >> 5;        // 0..3
    const int h2l  = wave >> 1;       // 0..1 -> h2 = h2b + h2l
    const int w2b  = (wave & 1) * 16;
    const int m    = lane & 15;       // N (cout) for C/D; M-part index for A
    const int hi   = lane >> 4;       // half-wave: K pair select

    float bv = (m < COUT) ? bias[m] : 0.0f;
    v8f acc = { bv, bv, bv, bv, bv, bv, bv, bv };

    // B fragment stream: lane reads pair [row][hi][m][0..1]; +64 floats per row.
    const v2f* __restrict__ bp = reinterpret_cast<const v2f*>(bpk) + (hi * 16 + m);
    // A fragment: 2 consecutive dwords at rowbase + m + 2*hi  (K = dw2 = 2*hi+{0,1})
    const int aoff = h2l * WROW + w2b + m + 2 * hi;

    for (int rw = 0; rw < CIN * 9; ++rw) {          // rw = ci*9 + dh1*3 + dw1
        const int rbase = rw * (H2E * WROW) + aoff;
        #pragma unroll
        for (int dh2 = 0; dh2 < 3; ++dh2) {
            v2f a;
            a.x = raw[rbase + dh2 * WROW];
            a.y = raw[rbase + dh2 * WROW + 1];
            v2f bw = *bp;
            acc = __builtin_amdgcn_wmma_f32_16x16x4_f32(
                      /*neg_a=*/false, a, /*neg_b=*/false, bw,
                      /*c_mod=*/(short)0, acc, /*reuse_a=*/false, /*reuse_b=*/false);
            bp += 32;                                // next K-row (64 floats)
        }
    }

    // ---- Store: lane N = m (cout), VGPR v -> w2 = w2b + 8*hi + v (contiguous) ----
    if (m < COUT) {
        size_t o = (((((size_t)n * COUT + m) * SDIM + h1) * SDIM + w1) * SDIM
                    + (h2b + h2l)) * SDIM + (w2b + 8 * hi);
        float4 s0, s1;
        s0.x = fmaxf(acc[0], 0.0f);  s0.y = fmaxf(acc[1], 0.0f);
        s0.z = fmaxf(acc[2], 0.0f);  s0.w = fmaxf(acc[3], 0.0f);
        s1.x = fmaxf(acc[4], 0.0f);  s1.y = fmaxf(acc[5], 0.0f);
        s1.z = fmaxf(acc[6], 0.0f);  s1.w = fmaxf(acc[7], 0.0f);
        *reinterpret_cast<float4*>(out + o)     = s0;
        *reinterpret_cast<float4*>(out + o + 4) = s1;
    }
}

// ---------------------------------------------------------------------------
// Layer 3: cout=1 (WMMA N-util would be 1/16) -> VALU FMA. ReLU, then write
// (pass 0) or accumulate (pass 1) into d_out. ~4% of total FLOPs per pass.
// ---------------------------------------------------------------------------
__global__ __launch_bounds__(256)
void conv4d_out_kernel(const float* __restrict__ h,    // [4][COUT][32^4]
                       const float* __restrict__ w3,   // [1][COUT][81]
                       const float* __restrict__ b3,
                       float* __restrict__ out, int permuted, int accum)
{
    size_t gid = (size_t)blockIdx.x * 256 + threadIdx.x;   // 4*32^4 threads
    int w2 = gid & 31;
    int h2 = (gid >> 5) & 31;
    int w1 = (gid >> 10) & 31;
    int h1 = (gid >> 15) & 31;
    int n  = (int)(gid >> 20);

    float acc = b3[0];
    for (int ci = 0; ci < COUT; ++ci) {
        const float* hp = h + (((size_t)n * COUT + ci) << 20);
        const float* wp = w3 + ci * 81;
        for (int a = 0; a < 3; ++a) {
            int h1g = h1 + a - 1; if ((unsigned)h1g >= 32u) continue;
            for (int b = 0; b < 3; ++b) {
                int w1g = w1 + b - 1; if ((unsigned)w1g >= 32u) continue;
                for (int c = 0; c < 3; ++c) {
                    int h2g = h2 + c - 1; if ((unsigned)h2g >= 32u) continue;
                    const float* row = hp + ((((size_t)h1g * 32 + w1g) * 32 + h2g) << 5);
                    #pragma unroll
                    for (int d = 0; d < 3; ++d) {
                        int w2g = w2 + d - 1;
                        if ((unsigned)w2g < 32u) {
                            int t = permuted ? (c * 27 + d * 9 + a * 3 + b)
                                             : (a * 27 + b * 9 + c * 3 + d);
                            acc = fmaf(row[w2g], wp[t], acc);
                        }
                    }
                }
            }
        }
    }
    acc = fmaxf(acc, 0.0f);
    if (accum) out[gid] += acc;
    else       out[gid]  = acc;
}

// ---------------------------------------------------------------------------
// Host: pack weights (both permutation variants), then two full passes:
//   pass0: relu(conv(x;W1)) -> h1 -> h2 -> out  (write)
//   pass1: same with tap-permuted W (== P(net(P(x)))) -> out (accumulate)
// ---------------------------------------------------------------------------
extern "C" void kernel_launch(void* const* d_in, const int* in_sizes, int n_in,
                              void* d_out, int out_size, void* d_ws, size_t ws_size,
                              hipStream_t stream)
{
    const float* x  = (const float*)d_in[0];
    const float* W1 = (const float*)d_in[1];
    const float* b1 = (const float*)d_in[2];
    const float* W2 = (const float*)d_in[3];
    const float* b2 = (const float*)d_in[4];
    const float* W3 = (const float*)d_in[5];
    const float* b3 = (const float*)d_in[6];
    float* out = (float*)d_out;

    const size_t HVOL = (size_t)4 * COUT * SDIM * SDIM * SDIM * SDIM; // 41,943,040 fl
    float* ws   = (float*)d_ws;
    float* h1   = ws;
    float* h2   = ws + HVOL;
    float* pk1  = ws + 2 * HVOL;          //  27*64 floats
    float* pk1p = pk1 + 27 * 64;
    float* pk2  = pk1p + 27 * 64;         // 270*64 floats
    float* pk2p = pk2 + 270 * 64;

    pack_weights_kernel<<<(27 * 64 + 255) / 256, 256, 0, stream>>>(W1, pk1, 1, 0);
    pack_weights_kernel<<<(27 * 64 + 255) / 256, 256, 0, stream>>>(W1, pk1p, 1, 1);
    pack_weights_kernel<<<(270 * 64 + 255) / 256, 256, 0, stream>>>(W2, pk2, 10, 0);
    pack_weights_kernel<<<(270 * 64 + 255) / 256, 256, 0, stream>>>(W2, pk2p, 10, 1);

    const int convGrid = 4 * 32 * 32 * 16;   // (n, h1, w1, h2/2) = 65536 blocks
    const int outGrid  = (4 << 20) / 256;    // 16384 blocks

    for (int pass = 0; pass < 2; ++pass) {
        conv4d_wmma_kernel<1><<<convGrid, 128, 0, stream>>>(
            x, pass ? pk1p : pk1, b1, h1);
        conv4d_wmma_kernel<10><<<convGrid, 128, 0, stream>>>(
            h1, pass ? pk2p : pk2, b2, h2);
        conv4d_out_kernel<<<outGrid, 256, 0, stream>>>(
            h2, W3, b3, out, pass, pass);
    }
}